// GCNNet_68856915689564
// MI455X (gfx1250) — compile-verified
//
#include <hip/hip_runtime.h>
#include <hip/hip_bf16.h>
#include <math.h>

typedef __attribute__((ext_vector_type(2))) float v2f;
typedef __attribute__((ext_vector_type(8))) float v8f;

#define EPS 1e-5f

__device__ __forceinline__ void gAtomicAdd(float* p, float v) {
    // lowers to global_atomic_add_f32 (no CAS loop)
    unsafeAtomicAdd(p, v);
}

// ---------------------------------------------------------------------------
// Degree count: cnt[dst] += 1 per edge
// ---------------------------------------------------------------------------
__global__ void count_edges_kernel(const int* __restrict__ ei, float* __restrict__ cnt, int E) {
    int e = blockIdx.x * blockDim.x + threadIdx.x;
    if (e >= E) return;
    int dst = ei[E + e];
    gAtomicAdd(&cnt[dst], 1.0f);
}

// ---------------------------------------------------------------------------
// Edge gather: agg[dst] += bn(h[src])   (BN optional; agg stride is 12)
// ---------------------------------------------------------------------------
template <int CIN, bool APPLY_BN>
__global__ void edge_gather_kernel(const int* __restrict__ ei,
                                   const float* __restrict__ hin,
                                   const float* __restrict__ bnscale,
                                   const float* __restrict__ bnshift,
                                   float* __restrict__ agg, int E) {
    int e = blockIdx.x * blockDim.x + threadIdx.x;
    if (e >= E) return;
    int src = ei[e];
    int dst = ei[E + e];
    const float* hrow = hin + (size_t)src * CIN;
    float v[CIN];
    if (CIN == 12) {
        const float4* h4 = (const float4*)hrow;  // 16B aligned: row stride 48B
        float4 r0 = h4[0], r1 = h4[1], r2 = h4[2];
        v[0] = r0.x; v[1] = r0.y; v[2] = r0.z; v[3] = r0.w;
        v[4] = r1.x; v[5] = r1.y; v[6] = r1.z; v[7] = r1.w;
        v[8] = r2.x; v[9] = r2.y; v[10] = r2.z; v[11] = r2.w;
    } else {
#pragma unroll
        for (int c = 0; c < CIN; ++c) v[c] = hrow[c];
    }
    if (APPLY_BN) {
#pragma unroll
        for (int c = 0; c < CIN; ++c) v[c] = fmaf(v[c], bnscale[c], bnshift[c]);
    }
    float* ag = agg + (size_t)dst * 12;
#pragma unroll
    for (int c = 0; c < CIN; ++c) gAtomicAdd(&ag[c], v[c]);
}

// ---------------------------------------------------------------------------
// SAGE node transform via V_WMMA_F32_16X16X4_F32:
//   out = relu( (agg/deg) @ Wl^T + bn(h) @ Wr^T + b )
// One wave (32 lanes) handles a 16-node tile; 8 waves per block.
// A layout (16x4 f32, 2 VGPR): lanes 0-15 -> M=lane, K={0,1}; lanes 16-31 -> K={2,3}
// B layout (4x16 f32, 2 VGPR): lanes 0-15 -> N=lane, K={0,1}; lanes 16-31 -> K={2,3}
// C/D (16x16 f32, 8 VGPR): vgpr r, lanes 0-15 -> M=r, N=lane; lanes 16-31 -> M=r+8
// ---------------------------------------------------------------------------
template <int CIN, bool APPLY_BN, bool ACC_STATS>
__launch_bounds__(256)
__global__ void sage_node_kernel(const float* __restrict__ hin,
                                 const float* __restrict__ agg,
                                 const float* __restrict__ cnt,
                                 const float* __restrict__ Wl,
                                 const float* __restrict__ Wr,
                                 const float* __restrict__ bias,
                                 const float* __restrict__ bnscale,
                                 const float* __restrict__ bnshift,
                                 float* __restrict__ hout,
                                 float* __restrict__ sums,
                                 float* __restrict__ sumsq) {
    constexpr int NCH = (CIN + 3) / 4;
    const int lane = threadIdx.x & 31;
    const int wave = threadIdx.x >> 5;
    const int tile = blockIdx.x * 8 + wave;
    const int m = lane & 15;       // N column / M row index within tile
    const int kh = lane >> 4;      // K half select (0 -> K 0,1 ; 1 -> K 2,3)
    const int rowbase = tile * 16;

    // ---- B operands (weights, transposed on the fly; zero-pad cols >= 12) ----
    v2f bl[NCH], br[NCH];
#pragma unroll
    for (int c = 0; c < NCH; ++c) {
        int k = 4 * c + 2 * kh;
        float w0 = 0.f, w1 = 0.f, u0 = 0.f, u1 = 0.f;
        if (m < 12) {
            if (k < CIN)     { w0 = Wl[m * CIN + k];     u0 = Wr[m * CIN + k]; }
            if (k + 1 < CIN) { w1 = Wl[m * CIN + k + 1]; u1 = Wr[m * CIN + k + 1]; }
        }
        bl[c].x = w0; bl[c].y = w1;
        br[c].x = u0; br[c].y = u1;
    }

    // ---- A operands: bn(h) rows and mean-aggregated rows ----
    const int node = rowbase + m;
    const float* hrow = hin + (size_t)node * CIN;
    const float* arow = agg + (size_t)node * 12;
    float inv = 1.0f / fmaxf(cnt[node], 1.0f);

    v2f ah[NCH], aa[NCH];
    if (CIN == 12) {
        // pairs (k, k+1) are 8B aligned: b64 loads
        const float2* h2 = (const float2*)hrow;
        const float2* a2 = (const float2*)arow;
#pragma unroll
        for (int c = 0; c < NCH; ++c) {
            int k = 4 * c + 2 * kh;
            float2 hp = h2[2 * c + kh];
            float2 ap = a2[2 * c + kh];
            float h0 = hp.x, h1 = hp.y;
            if (APPLY_BN) {
                h0 = fmaf(h0, bnscale[k], bnshift[k]);
                h1 = fmaf(h1, bnscale[k + 1], bnshift[k + 1]);
            }
            ah[c].x = h0;          ah[c].y = h1;
            aa[c].x = ap.x * inv;  aa[c].y = ap.y * inv;
        }
    } else {
#pragma unroll
        for (int c = 0; c < NCH; ++c) {
            int k = 4 * c + 2 * kh;
            float h0 = (k < CIN) ? hrow[k] : 0.f;
            float h1 = (k + 1 < CIN) ? hrow[k + 1] : 0.f;
            float a0 = (k < CIN) ? arow[k] * inv : 0.f;
            float a1 = (k + 1 < CIN) ? arow[k + 1] * inv : 0.f;
            ah[c].x = h0; ah[c].y = h1;
            aa[c].x = a0; aa[c].y = a1;
        }
    }

    // ---- C init with bias broadcast along M ----
    float bv = (m < 12) ? bias[m] : 0.f;
    v8f d;
#pragma unroll
    for (int r = 0; r < 8; ++r) d[r] = bv;

    // ---- accumulate both products with f32 WMMA ----
#pragma unroll
    for (int c = 0; c < NCH; ++c) {
        d = __builtin_amdgcn_wmma_f32_16x16x4_f32(false, aa[c], false, bl[c],
                                                  (short)0, d, false, false);
        d = __builtin_amdgcn_wmma_f32_16x16x4_f32(false, ah[c], false, br[c],
                                                  (short)0, d, false, false);
    }

    // ---- ReLU + per-channel stats (branch-free) ----
    float vr[8];
    float s = 0.f, q = 0.f;
#pragma unroll
    for (int r = 0; r < 8; ++r) {
        float v = fmaxf(d[r], 0.f);
        vr[r] = v;
        s += v;
        q += v * v;
    }

    // ---- single exec-masked store clause; immediate offsets r*48B ----
    if (m < 12) {
        float* op = hout + (size_t)(rowbase + 8 * kh) * 12 + m;
#pragma unroll
        for (int r = 0; r < 8; ++r) op[r * 12] = vr[r];
    }

    if (ACC_STATS) {
        __shared__ float ls[24];
        if (threadIdx.x < 24) ls[threadIdx.x] = 0.f;
        __syncthreads();
        if (m < 12) {
            atomicAdd(&ls[m], s);        // ds_add_f32
            atomicAdd(&ls[12 + m], q);
        }
        __syncthreads();
        if (threadIdx.x < 12) gAtomicAdd(&sums[threadIdx.x], ls[threadIdx.x]);
        else if (threadIdx.x < 24) gAtomicAdd(&sumsq[threadIdx.x - 12], ls[threadIdx.x]);
    }
}

// ---------------------------------------------------------------------------
// Fold sums into BN scale/shift; re-zero stats for the next accumulation.
// ---------------------------------------------------------------------------
__global__ void bn_finalize_kernel(const float* __restrict__ g, const float* __restrict__ b,
                                   float* sums, float* sumsq,
                                   float* scale, float* shift, float invn, int C) {
    int t = threadIdx.x;
    if (t < C) {
        float m = sums[t] * invn;
        float v = sumsq[t] * invn - m * m;
        float sc = g[t] * rsqrtf(fmaxf(v, 0.f) + EPS);
        scale[t] = sc;
        shift[t] = b[t] - m * sc;
        sums[t] = 0.f;
        sumsq[t] = 0.f;
    }
}

// ---------------------------------------------------------------------------
// Head: z = (relu(bn(in)))@W^T + b  (BN optional), accumulate z stats.
// ---------------------------------------------------------------------------
template <bool APPLY_BN>
__global__ void head_lin_kernel(const float* __restrict__ in,
                                const float* __restrict__ W,
                                const float* __restrict__ b,
                                const float* __restrict__ scale,
                                const float* __restrict__ shift,
                                float* __restrict__ out,
                                float* sums, float* sumsq, int rows) {
    __shared__ float ls[12];
    if (threadIdx.x < 12) ls[threadIdx.x] = 0.f;
    __syncthreads();
    int r = blockIdx.x * blockDim.x + threadIdx.x;
    if (r < rows) {
        const float* row = in + (size_t)r * 6;
        float a[6], z[6];
#pragma unroll
        for (int c = 0; c < 6; ++c) {
            float v = row[c];
            if (APPLY_BN) v = fmaxf(fmaf(v, scale[c], shift[c]), 0.f);
            a[c] = v;
        }
#pragma unroll
        for (int o = 0; o < 6; ++o) {
            float acc = b[o];
#pragma unroll
            for (int c = 0; c < 6; ++c) acc = fmaf(W[o * 6 + c], a[c], acc);
            z[o] = acc;
            out[(size_t)r * 6 + o] = acc;
        }
#pragma unroll
        for (int c = 0; c < 6; ++c) {
            atomicAdd(&ls[c], z[c]);
            atomicAdd(&ls[6 + c], z[c] * z[c]);
        }
    }
    __syncthreads();
    if (threadIdx.x < 6) gAtomicAdd(&sums[threadIdx.x], ls[threadIdx.x]);
    else if (threadIdx.x < 12) gAtomicAdd(&sumsq[threadIdx.x - 6], ls[threadIdx.x]);
}

__global__ void head_out_kernel(const float* __restrict__ in,
                                const float* __restrict__ scale,
                                const float* __restrict__ shift,
                                const float* __restrict__ oW,
                                const float* __restrict__ ob,
                                float* __restrict__ out, int rows) {
    int r = blockIdx.x * blockDim.x + threadIdx.x;
    if (r >= rows) return;
    const float* row = in + (size_t)r * 6;
    float acc = ob[0];
#pragma unroll
    for (int c = 0; c < 6; ++c) {
        float a = fmaxf(fmaf(row[c], scale[c], shift[c]), 0.f);
        acc = fmaf(oW[c], a, acc);
    }
    out[r] = 1.0f / (1.0f + expf(-acc));
}

// ---------------------------------------------------------------------------
extern "C" void kernel_launch(void* const* d_in, const int* in_sizes, int n_in,
                              void* d_out, int out_size, void* d_ws, size_t ws_size,
                              hipStream_t stream) {
    const float* x      = (const float*)d_in[0];
    const int*   ei     = (const int*)d_in[1];
    const float* Wl1    = (const float*)d_in[2];
    const float* Wr1    = (const float*)d_in[3];
    const float* b1     = (const float*)d_in[4];
    const float* Wl     = (const float*)d_in[5];
    const float* Wr     = (const float*)d_in[6];
    const float* bvec   = (const float*)d_in[7];
    const float* bn_g   = (const float*)d_in[8];
    const float* bn_b   = (const float*)d_in[9];
    const float* lin_W  = (const float*)d_in[10];
    const float* lin_b  = (const float*)d_in[11];
    const float* bn6_g  = (const float*)d_in[12];
    const float* bn6_b  = (const float*)d_in[13];
    const float* out_W  = (const float*)d_in[14];
    const float* out_b  = (const float*)d_in[15];
    float* out = (float*)d_out;

    const int N = in_sizes[0] / 3;
    const int E = in_sizes[1] / 2;

    float* ws   = (float*)d_ws;
    float* cnt  = ws;                    // N
    float* agg  = ws + (size_t)N;        // 12N
    float* hA   = ws + (size_t)13 * N;   // 12N
    float* hB   = ws + (size_t)25 * N;   // 12N
    float* smp  = ws + (size_t)37 * N;   // small params
    float* sums = smp;                   // 16
    float* sumq = smp + 16;              // 16
    float* bnsc = smp + 32;              // 16
    float* bnsh = smp + 48;              // 16

    const int TB = 256;
    const int egrid = (E + TB - 1) / TB;
    const int ngrid = (N + 127) / 128;        // 8 waves x 16 nodes per block
    const int rows  = 2 * N;
    const int hgrid = (rows + TB - 1) / TB;

    hipMemsetAsync(cnt, 0, (size_t)N * sizeof(float), stream);
    hipMemsetAsync(smp, 0, 64 * sizeof(float), stream);
    count_edges_kernel<<<egrid, TB, 0, stream>>>(ei, cnt, E);

    // ---- conv1: SAGE(3 -> 12) + ReLU, accumulate stats for BN1 ----
    hipMemsetAsync(agg, 0, (size_t)12 * N * sizeof(float), stream);
    edge_gather_kernel<3, false><<<egrid, TB, 0, stream>>>(ei, x, bnsc, bnsh, agg, E);
    sage_node_kernel<3, false, true><<<ngrid, TB, 0, stream>>>(
        x, agg, cnt, Wl1, Wr1, b1, bnsc, bnsh, hA, sums, sumq);

    // ---- conv2..conv8: BN then SAGE(12 -> 12) + ReLU ----
    for (int i = 0; i < 7; ++i) {
        bn_finalize_kernel<<<1, 32, 0, stream>>>(bn_g + i * 12, bn_b + i * 12,
                                                 sums, sumq, bnsc, bnsh, 1.0f / N, 12);
        hipMemsetAsync(agg, 0, (size_t)12 * N * sizeof(float), stream);
        const float* hin = (i & 1) ? hB : hA;
        float* hout      = (i & 1) ? hA : hB;
        edge_gather_kernel<12, true><<<egrid, TB, 0, stream>>>(ei, hin, bnsc, bnsh, agg, E);
        if (i < 6) {
            sage_node_kernel<12, true, true><<<ngrid, TB, 0, stream>>>(
                hin, agg, cnt, Wl + i * 144, Wr + i * 144, bvec + i * 12,
                bnsc, bnsh, hout, sums, sumq);
        } else {
            sage_node_kernel<12, true, false><<<ngrid, TB, 0, stream>>>(
                hin, agg, cnt, Wl + i * 144, Wr + i * 144, bvec + i * 12,
                bnsc, bnsh, hout, sums, sumq);
        }
    }
    // h8 lives in hB, viewed as [2N, 6]

    // ---- head: 3x (Linear(6,6) -> BN -> ReLU), then Linear(6,1) + sigmoid ----
    head_lin_kernel<false><<<hgrid, TB, 0, stream>>>(hB, lin_W, lin_b, bnsc, bnsh,
                                                     agg, sums, sumq, rows);   // z1 -> agg
    bn_finalize_kernel<<<1, 32, 0, stream>>>(bn6_g, bn6_b, sums, sumq, bnsc, bnsh,
                                             1.0f / rows, 6);
    head_lin_kernel<true><<<hgrid, TB, 0, stream>>>(agg, lin_W, lin_b, bnsc, bnsh,
                                                    hA, sums, sumq, rows);     // z2 -> hA
    bn_finalize_kernel<<<1, 32, 0, stream>>>(bn6_g, bn6_b, sums, sumq, bnsc, bnsh,
                                             1.0f / rows, 6);
    head_lin_kernel<true><<<hgrid, TB, 0, stream>>>(hA, lin_W, lin_b, bnsc, bnsh,
                                                    hB, sums, sumq, rows);     // z3 -> hB
    bn_finalize_kernel<<<1, 32, 0, stream>>>(bn6_g, bn6_b, sums, sumq, bnsc, bnsh,
                                             1.0f / rows, 6);
    head_out_kernel<<<hgrid, TB, 0, stream>>>(hB, bnsc, bnsh, out_W, out_b, out, rows);
}